// MGO_57767310131621
// MI455X (gfx1250) — compile-verified
//
#include <hip/hip_runtime.h>
#include <cstddef>

// ---------------------------------------------------------------------------
// CDNA5 / gfx1250 WMMA f32 16x16x4 helpers (wave32).
// A: 16x4  -> v2f per lane: lane l (<16): A[l, 0..1]; lane l (>=16): A[l-16, 2..3]
// B: 4x16  -> v2f per lane: lane l (<16): B[0..1, l]; lane l (>=16): B[2..3, l-16]
// C/D 16x16 f32 -> v8f: vgpr r: lanes0-15 C[r, l], lanes16-31 C[r+8, l-16]
// ---------------------------------------------------------------------------
typedef __attribute__((ext_vector_type(2))) float v2f;
typedef __attribute__((ext_vector_type(8))) float v8f;

__device__ inline v8f wmma_f32_k4(v2f a, v2f b, v8f c) {
  return __builtin_amdgcn_wmma_f32_16x16x4_f32(false, a, false, b, (short)0, c,
                                               false, false);
}
__device__ inline v8f v8zero() {
  v8f z = {0.f, 0.f, 0.f, 0.f, 0.f, 0.f, 0.f, 0.f};
  return z;
}
__device__ inline float selu_f(float x) {
  const float sc = 1.0507009873554805f, al = 1.6732632423543772f;
  return x > 0.f ? sc * x : sc * al * (expf(x) - 1.f);
}

// ------------------------------ build layer-0 data -------------------------
__global__ void build_data_k(const float* __restrict__ lhs,
                             const float* __restrict__ rhs,
                             const float* __restrict__ master,
                             float* __restrict__ data) {
  int idx = blockIdx.x * 256 + threadIdx.x;  // B(8) * NP(400) * D(64)
  if (idx >= 8 * 400 * 64) return;
  int d = idx & 63;
  int i = (idx >> 6) % 400;
  int b = idx / (400 * 64);
  float v;
  if (i < 192)       v = lhs[((size_t)b * 192 + i) * 64 + d];
  else if (i < 384)  v = rhs[((size_t)b * 192 + (i - 192)) * 64 + d];
  else if (i == 384) v = master[d];
  else               v = 0.f;
  data[idx] = v;
}

// ------------------------------ small row utilities ------------------------
__global__ void copy_row_k(const float* __restrict__ src, float* __restrict__ dst,
                           int srcB, int srcRow, int dstB, int dstRow) {
  int b = blockIdx.x, d = threadIdx.x;
  dst[((size_t)b * dstB + dstRow) * 64 + d] =
      src[((size_t)b * srcB + srcRow) * 64 + d];
}
__global__ void zero_rows_k(float* __restrict__ dst, int B_, int row0) {
  int b = blockIdx.y, r = row0 + blockIdx.x, d = threadIdx.x;
  dst[((size_t)b * B_ + r) * 64 + d] = 0.f;
}

// ------------------- partitioned transform: x@W1+b1 / x@W2+b2 --------------
// One wave per 16 flattened (b,i) rows of a partition; WMMA GEMM K=64.
__global__ __launch_bounds__(32) void transform_k(
    const float* __restrict__ data, const float* __restrict__ W1,
    const float* __restrict__ b1, const float* __restrict__ W2,
    const float* __restrict__ b2, float* __restrict__ xt, int Ls, int Rs,
    int NP) {
  int lane = threadIdx.x, half = lane >> 4, l16 = lane & 15;
  int t = blockIdx.x, T1 = (8 * Ls) >> 4;
  const float *W, *bias;
  int fr = t * 16 + l16, b, i;
  if (t < T1) { W = W1; bias = b1; b = fr / Ls; i = fr % Ls; }
  else { int f2 = fr - 8 * Ls; W = W2; bias = b2; b = f2 / Rs; i = Ls + f2 % Rs; }
  const float* arow = data + ((size_t)b * NP + i) * 64;
  v8f acc[4];
#pragma unroll
  for (int n = 0; n < 4; n++) acc[n] = v8zero();
#pragma unroll
  for (int k = 0; k < 16; k++) {
    int dr = 4 * k + 2 * half;
    v2f a = *(const v2f*)(arow + dr);
#pragma unroll
    for (int n = 0; n < 4; n++) {
      int o = n * 16 + l16;
      v2f bb;
      bb[0] = W[(size_t)(dr + 0) * 64 + o];
      bb[1] = W[(size_t)(dr + 1) * 64 + o];
      acc[n] = wmma_f32_k4(a, bb, acc[n]);
    }
  }
#pragma unroll
  for (int n = 0; n < 4; n++) {
    int o = n * 16 + l16;
    float bv = bias[o];
#pragma unroll
    for (int r = 0; r < 8; r++) {
      int frow = t * 16 + r + 8 * half, b2_, i2_;
      if (t < T1) { b2_ = frow / Ls; i2_ = frow % Ls; }
      else { int f2 = frow - 8 * Ls; b2_ = f2 / Rs; i2_ = Ls + f2 % Rs; }
      xt[((size_t)b2_ * NP + i2_) * 64 + o] = acc[n][r] + bv;
    }
  }
}

// ---------------- alpha: per (b,i) chunk of j-tiles, WMMA GEMM -------------
// pre[j,o] = sum_d (x_i[d]*x_j[d]) * W[d,o];  alpha[j] = sum_o tanh(pre+ba)*a_sel
// W (64x64) is cached in 128 VGPRs per wave and amortized over ACHUNK j-tiles;
// steady-state per tile: 16 xj loads + 64 v_wmma_f32_16x16x4_f32 + epilogue.
#define ACHUNK 5

__global__ __launch_bounds__(32) void alpha_k(
    const float* __restrict__ xt, const float* __restrict__ Wa,
    const float* __restrict__ ba, const float* __restrict__ a11,
    const float* __restrict__ a12, const float* __restrict__ a22,
    const float* __restrict__ WaM, const float* __restrict__ baM,
    const float* __restrict__ aM, float* __restrict__ alpha, int Ls, int Rs,
    int N, int NP) {
  int lane = threadIdx.x, half = lane >> 4, l16 = lane & 15;
  int i = blockIdx.y, b = blockIdx.z;
  int ntiles = NP >> 4;
  int tile0 = blockIdx.x * ACHUNK;
  bool isM = (i == N - 1);
  const float* W = isM ? WaM : Wa;
  const float* bias = isM ? baM : ba;

  // Cache the whole 64x64 W in registers (this wave's lane-slices).
  v2f Wr[16][4];
#pragma unroll
  for (int k = 0; k < 16; k++) {
    int dr = 4 * k + 2 * half;
#pragma unroll
    for (int n = 0; n < 4; n++) {
      int o = n * 16 + l16;
      Wr[k][n][0] = W[(size_t)(dr + 0) * 64 + o];
      Wr[k][n][1] = W[(size_t)(dr + 1) * 64 + o];
    }
  }
  // Cache x_i slices.
  const float* xi_p = xt + ((size_t)b * NP + i) * 64;
  v2f xi[16];
#pragma unroll
  for (int k = 0; k < 16; k++)
    xi[k] = *(const v2f*)(xi_p + 4 * k + 2 * half);
  // Epilogue constants (per o = n*16 + l16).
  float bo_[4], s1_[4], s2_[4];
#pragma unroll
  for (int n = 0; n < 4; n++) {
    int o = n * 16 + l16;
    bo_[n] = bias[o];
    if (isM) { s1_[n] = aM[o]; s2_[n] = s1_[n]; }
    else if (i < Ls) { s1_[n] = a11[o]; s2_[n] = a12[o]; }
    else { s1_[n] = a12[o]; s2_[n] = a22[o]; }
  }

  for (int tt = 0; tt < ACHUNK; tt++) {
    int t = tile0 + tt;
    if (t >= ntiles) break;
    int jt = t * 16;
    int j = jt + l16;
    if (j > N - 1) j = N - 1;
    const float* xj_p = xt + ((size_t)b * NP + j) * 64;
    v8f acc[4];
#pragma unroll
    for (int n = 0; n < 4; n++) acc[n] = v8zero();
#pragma unroll
    for (int k = 0; k < 16; k++) {
      v2f c = *(const v2f*)(xj_p + 4 * k + 2 * half);
      v2f u;
      u[0] = xi[k][0] * c[0];
      u[1] = xi[k][1] * c[1];
#pragma unroll
      for (int n = 0; n < 4; n++) acc[n] = wmma_f32_k4(u, Wr[k][n], acc[n]);
    }
    float part[8] = {0, 0, 0, 0, 0, 0, 0, 0};
#pragma unroll
    for (int n = 0; n < 4; n++) {
#pragma unroll
      for (int r = 0; r < 8; r++) {
        int jj = jt + r + 8 * half;
        float sel;
        if (isM) sel = (jj < N) ? s1_[n] : 0.f;
        else sel = (jj < Ls) ? s1_[n] : ((jj < Ls + Rs) ? s2_[n] : 0.f);
        part[r] += tanhf(acc[n][r] + bo_[n]) * sel;
      }
    }
#pragma unroll
    for (int m = 1; m < 16; m <<= 1) {
#pragma unroll
      for (int r = 0; r < 8; r++) part[r] += __shfl_xor(part[r], m, 32);
    }
    if (l16 == 0) {
      float* row = alpha + ((size_t)b * N + i) * NP;
#pragma unroll
      for (int r = 0; r < 8; r++) row[jt + r + 8 * half] = part[r];
    }
  }
}

// ------------------------------ softmax over j -----------------------------
__global__ __launch_bounds__(32) void softmax_k(float* __restrict__ alpha, int N,
                                                int NP) {
  int i = blockIdx.x, b = blockIdx.y, lane = threadIdx.x;
  float* row = alpha + ((size_t)b * N + i) * NP;
  float m = -3.0e38f;
  for (int j = lane; j < N; j += 32) m = fmaxf(m, row[j]);
#pragma unroll
  for (int s = 16; s > 0; s >>= 1) m = fmaxf(m, __shfl_xor(m, s, 32));
  float sum = 0.f;
  for (int j = lane; j < N; j += 32) sum += expf(row[j] - m);
#pragma unroll
  for (int s = 16; s > 0; s >>= 1) sum += __shfl_xor(sum, s, 32);
  float inv = 1.f / sum;
  for (int j = lane; j < NP; j += 32)
    row[j] = (j < N) ? expf(row[j] - m) * inv : 0.f;
}

// ------------------------ weighted = attn @ x (WMMA) -----------------------
__global__ __launch_bounds__(32) void weighted_k(const float* __restrict__ attn,
                                                 const float* __restrict__ xt,
                                                 float* __restrict__ wgt, int N,
                                                 int NP) {
  int lane = threadIdx.x, half = lane >> 4, l16 = lane & 15;
  int it = blockIdx.x * 16, b = blockIdx.y;
  int ir = it + l16;
  if (ir > N - 1) ir = N - 1;
  const float* arow = attn + ((size_t)b * N + ir) * NP;
  const float* xb = xt + (size_t)b * NP * 64;
  v8f acc[4];
#pragma unroll
  for (int n = 0; n < 4; n++) acc[n] = v8zero();
  int ksteps = NP >> 2;
  for (int k = 0; k < ksteps; k++) {
    int j0 = 4 * k + 2 * half;
    v2f a = *(const v2f*)(arow + j0);
    const float* x0 = xb + (size_t)j0 * 64;
#pragma unroll
    for (int n = 0; n < 4; n++) {
      int o = n * 16 + l16;
      v2f bb;
      bb[0] = x0[o];
      bb[1] = x0[64 + o];
      acc[n] = wmma_f32_k4(a, bb, acc[n]);
    }
  }
#pragma unroll
  for (int n = 0; n < 4; n++) {
    int o = n * 16 + l16;
#pragma unroll
    for (int r = 0; r < 8; r++) {
      int i = it + r + 8 * half;
      if (i < N) wgt[((size_t)b * NP + i) * 64 + o] = acc[n][r];
    }
  }
}

// ---------------- out_main = weighted@Ww + x@Wo + bw + bo (WMMA) -----------
__global__ __launch_bounds__(32) void out_main_k(
    const float* __restrict__ wgt, const float* __restrict__ xt,
    const float* __restrict__ Ww, const float* __restrict__ bw,
    const float* __restrict__ Wo, const float* __restrict__ bo,
    float* __restrict__ outb, int N, int NP) {
  int lane = threadIdx.x, half = lane >> 4, l16 = lane & 15;
  int it = blockIdx.x * 16, b = blockIdx.y;
  int ir = it + l16;
  if (ir > N - 1) ir = N - 1;
  const float* wr = wgt + ((size_t)b * NP + ir) * 64;
  const float* xr = xt + ((size_t)b * NP + ir) * 64;
  v8f acc[4];
#pragma unroll
  for (int n = 0; n < 4; n++) acc[n] = v8zero();
#pragma unroll
  for (int k = 0; k < 16; k++) {
    int dr = 4 * k + 2 * half;
    v2f a1 = *(const v2f*)(wr + dr);
    v2f a2 = *(const v2f*)(xr + dr);
#pragma unroll
    for (int n = 0; n < 4; n++) {
      int o = n * 16 + l16;
      v2f b1v, b2v;
      b1v[0] = Ww[(size_t)(dr + 0) * 64 + o];
      b1v[1] = Ww[(size_t)(dr + 1) * 64 + o];
      acc[n] = wmma_f32_k4(a1, b1v, acc[n]);
      b2v[0] = Wo[(size_t)(dr + 0) * 64 + o];
      b2v[1] = Wo[(size_t)(dr + 1) * 64 + o];
      acc[n] = wmma_f32_k4(a2, b2v, acc[n]);
    }
  }
#pragma unroll
  for (int n = 0; n < 4; n++) {
    int o = n * 16 + l16;
    float bias = bw[o] + bo[o];
#pragma unroll
    for (int r = 0; r < 8; r++) {
      int i = it + r + 8 * half;
      if (i < N - 1) outb[((size_t)b * NP + i) * 64 + o] = acc[n][r] + bias;
    }
  }
}

// ------------------------------ master-row output --------------------------
__global__ void out_master_k(const float* __restrict__ wgt,
                             const float* __restrict__ xt,
                             const float* __restrict__ WwM,
                             const float* __restrict__ bwM,
                             const float* __restrict__ WoM,
                             const float* __restrict__ boM,
                             float* __restrict__ outb, int N, int NP) {
  int b = blockIdx.x, o = threadIdx.x;
  const float* wr = wgt + ((size_t)b * NP + (N - 1)) * 64;
  const float* xr = xt + ((size_t)b * NP + (N - 1)) * 64;
  float acc = bwM[o] + boM[o];
  for (int d = 0; d < 64; d++)
    acc += wr[d] * WwM[(size_t)d * 64 + o] + xr[d] * WoM[(size_t)d * 64 + o];
  outb[((size_t)b * NP + (N - 1)) * 64 + o] = acc;
}

// ------------------- layernorm stats over (b, i<N-1) per channel -----------
__global__ void stats_k(const float* __restrict__ outb, int N, int NP,
                        float* __restrict__ stats) {
  int d = blockIdx.x, t = threadIdx.x;
  __shared__ float s1[256], s2[256];
  int rows = 8 * (N - 1);
  float a = 0.f, q = 0.f;
  for (int rr = t; rr < rows; rr += 256) {
    int b = rr / (N - 1), i = rr % (N - 1);
    float v = outb[((size_t)b * NP + i) * 64 + d];
    a += v;
    q += v * v;
  }
  s1[t] = a;
  s2[t] = q;
  __syncthreads();
  for (int st = 128; st > 0; st >>= 1) {
    if (t < st) { s1[t] += s1[t + st]; s2[t] += s2[t + st]; }
    __syncthreads();
  }
  if (t == 0) {
    float mean = s1[0] / rows;
    float var = s2[0] / rows - mean * mean;
    stats[d] = mean;
    stats[64 + d] = rsqrtf(var + 1e-5f);
  }
}

__global__ void norm_k(float* __restrict__ outb, const float* __restrict__ stats,
                       const float* __restrict__ gamma,
                       const float* __restrict__ beta, int N, int NP) {
  int idx = blockIdx.x * 256 + threadIdx.x;
  int rows = 8 * (N - 1);
  if (idx >= rows * 64) return;
  int d = idx & 63, rr = idx >> 6;
  int b = rr / (N - 1), i = rr % (N - 1);
  size_t off = ((size_t)b * NP + i) * 64 + d;
  float v = outb[off];
  v = (v - stats[d]) * stats[64 + d] * gamma[d] + beta[d];
  outb[off] = selu_f(v);
}

// ------------------------------ pooling scores -----------------------------
__global__ void scores_k(const float* __restrict__ outb,
                         const float* __restrict__ pw1,
                         const float* __restrict__ pb1,
                         const float* __restrict__ pw2,
                         const float* __restrict__ pb2,
                         float* __restrict__ scores, int Ls, int NP) {
  int i = blockIdx.x, b = blockIdx.y, d = threadIdx.x;
  const float* pw = (i < Ls) ? pw1 : pw2;
  float p = outb[((size_t)b * NP + i) * 64 + d] * pw[d];
  __shared__ float s[64];
  s[d] = p;
  __syncthreads();
  for (int st = 32; st > 0; st >>= 1) {
    if (d < st) s[d] += s[d + st];
    __syncthreads();
  }
  if (d == 0) {
    float z = s[0] + ((i < Ls) ? pb1[0] : pb2[0]);
    scores[(size_t)b * 384 + i] = 1.f / (1.f + expf(-z));
  }
}

// ------------- top-k (descending, low-index ties) + gather*val -------------
__global__ void topk_gather_k(const float* __restrict__ src,
                              const float* __restrict__ scores,
                              float* __restrict__ dst, int n, int k, int srcB,
                              int seg0, int dstB, int dstoff) {
  int b = blockIdx.x, t = threadIdx.x;
  __shared__ float vals[256];
  __shared__ float rv[256];
  __shared__ int ri[256];
  __shared__ int selidx[144];
  __shared__ float selval[144];
  vals[t] = (t < n) ? scores[(size_t)b * 384 + seg0 + t] : -3.0e38f;
  __syncthreads();
  for (int r = 0; r < k; r++) {
    rv[t] = vals[t];
    ri[t] = t;
    __syncthreads();
    for (int s = 128; s > 0; s >>= 1) {
      if (t < s) {
        float v2 = rv[t + s];
        int i2 = ri[t + s];
        if (v2 > rv[t] || (v2 == rv[t] && i2 < ri[t])) { rv[t] = v2; ri[t] = i2; }
      }
      __syncthreads();
    }
    if (t == 0) {
      selidx[r] = ri[0];
      selval[r] = rv[0];
      vals[ri[0]] = -3.0e38f;
    }
    __syncthreads();
  }
  for (int e = t; e < k * 64; e += 256) {
    int r = e >> 6, d = e & 63;
    dst[((size_t)b * dstB + dstoff + r) * 64 + d] =
        src[((size_t)b * srcB + seg0 + selidx[r]) * 64 + d] * selval[r];
  }
}

// ---------------- branch max-combine + feature build + head ----------------
__global__ void combine_head_k(const float* __restrict__ fl0,
                               const float* __restrict__ fl1,
                               const float* __restrict__ fr0,
                               const float* __restrict__ fr1,
                               const float* __restrict__ fm0,
                               const float* __restrict__ fm1,
                               const float* __restrict__ hw,
                               const float* __restrict__ hb,
                               float* __restrict__ out) {
  int b = blockIdx.x, d = threadIdx.x;
  __shared__ float feat[320];
  float mx = -3.0e38f, sm = 0.f;
  for (int i = 0; i < 48; i++) {
    float v = fmaxf(fl0[((size_t)b * 48 + i) * 64 + d],
                    fl1[((size_t)b * 48 + i) * 64 + d]);
    mx = fmaxf(mx, v);
    sm += v;
  }
  feat[d] = mx;
  feat[64 + d] = sm / 48.f;
  mx = -3.0e38f;
  sm = 0.f;
  for (int i = 0; i < 93; i++) {
    float v = fmaxf(fr0[((size_t)b * 93 + i) * 64 + d],
                    fr1[((size_t)b * 93 + i) * 64 + d]);
    mx = fmaxf(mx, v);
    sm += v;
  }
  feat[128 + d] = mx;
  feat[192 + d] = sm / 93.f;
  feat[256 + d] = fmaxf(fm0[b * 64 + d], fm1[b * 64 + d]);
  __syncthreads();
  if (d < 2) {
    float acc = hb[d];
    for (int q = 0; q < 320; q++) acc += feat[q] * hw[q * 2 + d];
    out[b * 2 + d] = acc;
  }
}

// ===========================================================================
// Host orchestration
// ===========================================================================
enum {
  iW1, iB1, iW2, iB2, iWA, iBA, iWAM, iBAM, iA11, iA12, iA22, iAM,
  iWW, iBW, iWO, iBO, iWWM, iBWM, iWOM, iBOM, iGAM, iBET, iPW1, iPB1, iPW2, iPB2
};
struct LP { const float* a[26]; };
static LP lp_from(void* const* din, int b0) {
  LP l;
  for (int i = 0; i < 26; i++) l.a[i] = (const float*)din[b0 + i];
  return l;
}

static void run_layer(const LP& lp, int Ls, int Rs, int N, int NP, float* data,
                      float* xt, float* alpha, float* wgt, float* outb,
                      float* stats, float* scores, hipStream_t stream) {
  int T1 = (8 * Ls) / 16, T2 = (8 * Rs) / 16;
  transform_k<<<T1 + T2, 32, 0, stream>>>(data, lp.a[iW1], lp.a[iB1], lp.a[iW2],
                                          lp.a[iB2], xt, Ls, Rs, NP);
  copy_row_k<<<8, 64, 0, stream>>>(data, xt, NP, N - 1, NP, N - 1);
  zero_rows_k<<<dim3(NP - N, 8), 64, 0, stream>>>(xt, NP, N);
  int ntiles = NP / 16;
  int nchunks = (ntiles + ACHUNK - 1) / ACHUNK;
  alpha_k<<<dim3(nchunks, N, 8), 32, 0, stream>>>(
      xt, lp.a[iWA], lp.a[iBA], lp.a[iA11], lp.a[iA12], lp.a[iA22], lp.a[iWAM],
      lp.a[iBAM], lp.a[iAM], alpha, Ls, Rs, N, NP);
  softmax_k<<<dim3(N, 8), 32, 0, stream>>>(alpha, N, NP);
  weighted_k<<<dim3(ntiles, 8), 32, 0, stream>>>(alpha, xt, wgt, N, NP);
  out_main_k<<<dim3(ntiles, 8), 32, 0, stream>>>(
      wgt, xt, lp.a[iWW], lp.a[iBW], lp.a[iWO], lp.a[iBO], outb, N, NP);
  out_master_k<<<8, 64, 0, stream>>>(wgt, xt, lp.a[iWWM], lp.a[iBWM],
                                     lp.a[iWOM], lp.a[iBOM], outb, N, NP);
  stats_k<<<64, 256, 0, stream>>>(outb, N, NP, stats);
  norm_k<<<(8 * (N - 1) * 64 + 255) / 256, 256, 0, stream>>>(
      outb, stats, lp.a[iGAM], lp.a[iBET], N, NP);
  scores_k<<<dim3(Ls + Rs, 8), 64, 0, stream>>>(
      outb, lp.a[iPW1], lp.a[iPB1], lp.a[iPW2], lp.a[iPB2], scores, Ls, NP);
}

extern "C" void kernel_launch(void* const* d_in, const int* in_sizes, int n_in,
                              void* d_out, int out_size, void* d_ws,
                              size_t ws_size, hipStream_t stream) {
  (void)in_sizes; (void)n_in; (void)out_size; (void)ws_size;
  const float* lhs = (const float*)d_in[0];
  const float* rhs = (const float*)d_in[1];
  const float* head_w = (const float*)d_in[108];
  const float* head_b = (const float*)d_in[109];
  float* ws = (float*)d_ws;

  const size_t OFF_DATA = 0;                        // 8*400*64
  const size_t OFF_XT = OFF_DATA + 204800;          // 8*400*64
  const size_t OFF_ALPHA = OFF_XT + 204800;         // 8*385*400
  const size_t OFF_WGT = OFF_ALPHA + 1232000;       // 8*400*64
  const size_t OFF_OUT = OFF_WGT + 204800;          // 8*400*64
  const size_t OFF_STATS = OFF_OUT + 204800;        // 128
  const size_t OFF_SC = OFF_STATS + 128;            // 8*384
  const size_t OFF_FL = OFF_SC + 3072;              // 2 * 8*48*64
  const size_t OFF_FR = OFF_FL + 49152;             // 2 * 8*93*64
  const size_t OFF_FM = OFF_FR + 95232;             // 2 * 8*64

  float* DATA = ws + OFF_DATA;
  float* XT = ws + OFF_XT;
  float* ALPHA = ws + OFF_ALPHA;
  float* WGT = ws + OFF_WGT;
  float* OUT = ws + OFF_OUT;
  float* STATS = ws + OFF_STATS;
  float* SC = ws + OFF_SC;
  float* FL = ws + OFF_FL;
  float* FR = ws + OFF_FR;
  float* FM = ws + OFF_FM;

  for (int br = 0; br < 2; ++br) {
    int mbase = (br == 0) ? 2 : 55;  // branch master param index
    const float* masterp = (const float*)d_in[mbase];
    LP l0 = lp_from(d_in, mbase + 1);
    LP l1 = lp_from(d_in, mbase + 27);

    build_data_k<<<(8 * 400 * 64 + 255) / 256, 256, 0, stream>>>(lhs, rhs,
                                                                 masterp, DATA);
    // layer 0: Ls=192, Rs=192, N=385, NP=400; pools k1=96, k2=134
    run_layer(l0, 192, 192, 385, 400, DATA, XT, ALPHA, WGT, OUT, STATS, SC,
              stream);
    topk_gather_k<<<8, 256, 0, stream>>>(OUT, SC, DATA, 192, 96, 400, 0, 240, 0);
    topk_gather_k<<<8, 256, 0, stream>>>(OUT, SC, DATA, 192, 134, 400, 192, 240,
                                         96);
    copy_row_k<<<8, 64, 0, stream>>>(OUT, DATA, 400, 384, 240, 230);
    zero_rows_k<<<dim3(9, 8), 64, 0, stream>>>(DATA, 240, 231);

    // layer 1: Ls=96, Rs=134, N=231, NP=240; pools k1=48, k2=93
    run_layer(l1, 96, 134, 231, 240, DATA, XT, ALPHA, WGT, OUT, STATS, SC,
              stream);
    float* FLb = FL + (size_t)br * 8 * 48 * 64;
    float* FRb = FR + (size_t)br * 8 * 93 * 64;
    float* FMb = FM + (size_t)br * 8 * 64;
    topk_gather_k<<<8, 256, 0, stream>>>(OUT, SC, FLb, 96, 48, 240, 0, 48, 0);
    topk_gather_k<<<8, 256, 0, stream>>>(OUT, SC, FRb, 134, 93, 240, 96, 93, 0);
    copy_row_k<<<8, 64, 0, stream>>>(OUT, FMb, 240, 230, 1, 0);
  }

  combine_head_k<<<8, 64, 0, stream>>>(FL, FL + 8 * 48 * 64, FR,
                                       FR + 8 * 93 * 64, FM, FM + 8 * 64,
                                       head_w, head_b, (float*)d_out);
}